// RGCNLayer_35854386987426
// MI455X (gfx1250) — compile-verified
//
#include <hip/hip_runtime.h>
#include <hip/hip_bf16.h>

typedef float v2f __attribute__((ext_vector_type(2)));
typedef float v8f __attribute__((ext_vector_type(8)));

// ---------------------------------------------------------------------------
// GEMM: C[M x 128] = A[M x 128] * B[128 x 128], f32, via V_WMMA_F32_16X16X4_F32
// Block = 256 threads = 8 waves. Wave w owns output columns [16w, 16w+16).
// Each wave preloads its 128x16 B slab into 32 v2f registers (64 VGPRs) and
// grid-strides over 16-row M tiles.
//
// WMMA f32 16x16x4 register layout (ISA 7.12.2):
//   A (16x4):  lanes 0-15 hold M=lane, K={0,1}; lanes 16-31 hold M=lane-16, K={2,3}
//   B (4x16):  VGPR0: lanes 0-15 = row K=0, lanes 16-31 = row K=2; VGPR1: K=1 / K=3
//   C (16x16): VGPR i: lanes 0-15 -> M=i, N=lane; lanes 16-31 -> M=i+8, N=lane-16
// ---------------------------------------------------------------------------
__global__ __launch_bounds__(256) void rgcn_gemm128(const float* __restrict__ A,
                                                    const float* __restrict__ B,
                                                    float* __restrict__ C,
                                                    int mTiles, int nRows) {
    const int lane  = threadIdx.x & 31;
    const int wave  = threadIdx.x >> 5;          // 0..7 -> which 16-col slab
    const int lcol  = lane & 15;
    const int col   = wave * 16 + lcol;
    const int koff  = (lane >> 4) << 1;          // 0 for lanes 0-15, 2 for 16-31

    // Preload B slab: Bv[kk] covers K = 4*kk .. 4*kk+3 for this wave's columns.
    v2f Bv[32];
#pragma unroll
    for (int kk = 0; kk < 32; ++kk) {
        const int k = kk * 4 + koff;
        Bv[kk].x = B[(size_t)k * 128 + col];
        Bv[kk].y = B[(size_t)(k + 1) * 128 + col];
    }

    for (int mt = blockIdx.x; mt < mTiles; mt += gridDim.x) {
        const int row  = mt * 16 + lcol;
        const int rowc = (row < nRows) ? row : (nRows - 1);   // clamp, OOB rows never stored
        const float* Arow = A + (size_t)rowc * 128 + koff;

        v8f acc0 = {};
        v8f acc1 = {};
#pragma unroll
        for (int kk = 0; kk < 32; kk += 2) {
            const float2 a0f = *(const float2*)(Arow + kk * 4);
            const float2 a1f = *(const float2*)(Arow + kk * 4 + 4);
            v2f a0; a0.x = a0f.x; a0.y = a0f.y;
            v2f a1; a1.x = a1f.x; a1.y = a1f.y;
            acc0 = __builtin_amdgcn_wmma_f32_16x16x4_f32(false, a0, false, Bv[kk],
                                                         (short)0, acc0, false, false);
            acc1 = __builtin_amdgcn_wmma_f32_16x16x4_f32(false, a1, false, Bv[kk + 1],
                                                         (short)0, acc1, false, false);
        }
        const v8f acc = acc0 + acc1;

#pragma unroll
        for (int i = 0; i < 8; ++i) {
            const int orow = mt * 16 + i + ((lane >> 4) << 3);
            if (orow < nRows) {
                C[(size_t)orow * 128 + col] = acc[i];
            }
        }
    }
}

// ---------------------------------------------------------------------------
// Per-edge scatter: out[dst] += xw_slab[src]  (128 f32 per edge, 1 wave/edge).
// rFilter >= 0: only edges whose type == rFilter, slab base = xw.
// rFilter <  0: slab base = xw + edge_type * slabStride (all relations resident).
// ---------------------------------------------------------------------------
__global__ __launch_bounds__(256) void rgcn_scatter(const float* __restrict__ xw,
                                                    const long long* __restrict__ src,
                                                    const long long* __restrict__ dst,
                                                    const long long* __restrict__ etype,
                                                    float* __restrict__ out,
                                                    int E, int rFilter, long long slabStride) {
    const int e = blockIdx.x * (blockDim.x >> 5) + (threadIdx.x >> 5);
    if (e >= E) return;
    const long long t = etype[e];
    const float* base;
    if (rFilter >= 0) {
        if ((int)t != rFilter) return;
        base = xw;
    } else {
        base = xw + t * slabStride;
    }
    const int lane = threadIdx.x & 31;
    const long long s = src[e];
    const long long d = dst[e];
    const float4 v = reinterpret_cast<const float4*>(base + s * 128)[lane];
    float* o = out + d * 128 + (size_t)lane * 4;
    atomicAdd(o + 0, v.x);
    atomicAdd(o + 1, v.y);
    atomicAdd(o + 2, v.z);
    atomicAdd(o + 3, v.w);
}

__global__ __launch_bounds__(256) void rgcn_zero(float* __restrict__ p, int n) {
    const int i = blockIdx.x * blockDim.x + threadIdx.x;
    if (i < n) p[i] = 0.0f;
}

__global__ __launch_bounds__(256) void rgcn_deg(const long long* __restrict__ dst,
                                                float* __restrict__ deg, int E) {
    const int e = blockIdx.x * blockDim.x + threadIdx.x;
    if (e < E) atomicAdd(&deg[(int)dst[e]], 1.0f);
}

__global__ __launch_bounds__(256) void rgcn_div(float* __restrict__ out,
                                                const float* __restrict__ deg, int total) {
    const int i = blockIdx.x * blockDim.x + threadIdx.x;
    if (i < total) {
        const float d = deg[i >> 7];       // OUT_DIM == 128
        out[i] *= __frcp_rn(fmaxf(d, 1.0f));
    }
}

extern "C" void kernel_launch(void* const* d_in, const int* in_sizes, int n_in,
                              void* d_out, int out_size, void* d_ws, size_t ws_size,
                              hipStream_t stream) {
    const float*     x  = (const float*)d_in[0];
    const long long* ei = (const long long*)d_in[1];   // [2, E] int64
    const long long* et = (const long long*)d_in[2];   // [E]    int64
    const float*     W  = (const float*)d_in[4];       // [R, 128, 128]
    const float*     W0 = (const float*)d_in[5];       // [128, 128]
    float*           out = (float*)d_out;

    const int IN  = 128;
    const int OUT = 128;
    const int N = in_sizes[0] / IN;
    const int E = in_sizes[2];
    const int R = in_sizes[4] / (IN * OUT);
    const int mTiles = (N + 15) / 16;

    const long long* src = ei;
    const long long* dst = ei + E;

    // Workspace layout: [deg: N f32, padded to 256B] [xw slab(s)]
    const size_t degPad    = (((size_t)N * sizeof(float)) + 255) & ~(size_t)255;
    const size_t slabElems = (size_t)N * OUT;
    const bool   big = ws_size >= degPad + (size_t)R * slabElems * sizeof(float);

    float* deg = (float*)d_ws;
    float* xw  = (float*)((char*)d_ws + degPad);

    const int gemmGrid    = 160;
    const int zeroGrid    = (N + 255) / 256;
    const int edgeGrid    = (E + 255) / 256;
    const int waveGrid    = (E + 7) / 8;          // 8 waves (edges) per 256-thread block
    const int divGrid     = ((int)((size_t)N * OUT) + 255) / 256;

    // Degree
    rgcn_zero<<<zeroGrid, 256, 0, stream>>>(deg, N);
    rgcn_deg<<<edgeGrid, 256, 0, stream>>>(dst, deg, E);

    // Root/self term: out = x @ W0 (fully overwrites d_out before atomics)
    rgcn_gemm128<<<gemmGrid, 256, 0, stream>>>(x, W0, out, mTiles, N);

    if (big) {
        for (int r = 0; r < R; ++r) {
            rgcn_gemm128<<<gemmGrid, 256, 0, stream>>>(x, W + (size_t)r * IN * OUT,
                                                       xw + (size_t)r * slabElems, mTiles, N);
        }
        rgcn_scatter<<<waveGrid, 256, 0, stream>>>(xw, src, dst, et, out, E,
                                                   -1, (long long)slabElems);
    } else {
        for (int r = 0; r < R; ++r) {
            rgcn_gemm128<<<gemmGrid, 256, 0, stream>>>(x, W + (size_t)r * IN * OUT,
                                                       xw, mTiles, N);
            rgcn_scatter<<<waveGrid, 256, 0, stream>>>(xw, src, dst, et, out, E,
                                                       r, 0);
        }
    }

    // Normalize by degree
    rgcn_div<<<divGrid, 256, 0, stream>>>(out, deg, (int)((size_t)N * OUT));
}